// GATv2Conv_39599598469259
// MI455X (gfx1250) — compile-verified
//
#include <hip/hip_runtime.h>

// GATv2Conv for MI455X (gfx1250, wave32).
// Phases:
//   0) zero out/emax/denom
//   1) WMMA fp32 GEMM: x_l = x@W_l, x_r = x@W_r   (V_WMMA_F32_16X16X4_F32)
//      weights staged into LDS via TDM (tensor_load_to_lds + s_wait_tensorcnt)
//   2) per-edge logits e = att . leaky(x_l[src]+x_r[dst]); atomicMax per dst
//   3) per-edge w = exp(e - max); denom[dst]+=w; out[dst]+=w*x_l[src] (atomics)
//   4) out = out/denom + bias

typedef __attribute__((ext_vector_type(2))) float v2f;
typedef __attribute__((ext_vector_type(8))) float v8f;

#define NEG_SLOPE 0.2f

#if __has_builtin(__builtin_amdgcn_tensor_load_to_lds)
#define USE_TDM 1
#else
#define USE_TDM 0
#endif

#if USE_TDM
typedef unsigned __attribute__((ext_vector_type(4))) v4u;
typedef int __attribute__((ext_vector_type(4))) v4i;
typedef int __attribute__((ext_vector_type(8))) v8i;

// 1-D TDM copy: 4096 fp32 (16 KB) from global -> LDS at byte offset lds_off.
// D# per CDNA5 ISA §8.3/8.4: group0 = {count=1 | lds_addr | global_addr | type=2},
// group1 = {data_size=4B, tensor_dim0=4096, tensor_dim1=1, tile_dim0=4096,
//           tile_dim1/2=0(unused), tensor_dim0_stride=4096}.
// This toolchain's builtin takes 6 args (extra int32x8 group before cpol).
__device__ __forceinline__ void tdm_load_w(const float* gsrc, unsigned lds_off) {
  unsigned long long ga = (unsigned long long)(const void*)gsrc;
  v4u g0 = {1u,                                              // count=1 (valid)
            lds_off,                                         // lds_addr[31:0]
            (unsigned)ga,                                    // global_addr[31:0]
            (unsigned)((ga >> 32) & 0x1FFFFFFu) | 0x80000000u};  // ga[56:32]|type=2
  v8i g1 = {(int)0x00020000,     // workgroup_mask=0, data_size=2 (4 bytes)
            (int)(4096u << 16),  // tensor_dim0[15:0]=4096 (bits 63:48)
            (int)(1u << 16),     // tensor_dim0[31:16]=0 | tensor_dim1[15:0]=1
            (int)(4096u << 16),  // tensor_dim1[31:16]=0 | tile_dim0=4096
            0,                   // tile_dim1=0, tile_dim2=0 (1-D tile)
            4096,                // tensor_dim0_stride[31:0]
            0, 0};               // stride0 hi / tensor_dim1_stride
  v4i z4 = {0, 0, 0, 0};         // groups 2/3 unused (<=2-D tensor)
  v8i z8 = {0, 0, 0, 0, 0, 0, 0, 0};
  __builtin_amdgcn_tensor_load_to_lds(g0, g1, z4, z4, z8, 0);
}
#endif

__device__ __forceinline__ unsigned f2ord(float f) {
  unsigned u = __float_as_uint(f);
  return (u & 0x80000000u) ? ~u : (u | 0x80000000u);
}
__device__ __forceinline__ float ord2f(unsigned u) {
  return (u & 0x80000000u) ? __uint_as_float(u & 0x7fffffffu)
                           : __uint_as_float(~u);
}

__global__ void k_zero_u32(unsigned* __restrict__ p, int n) {
  int i = blockIdx.x * 256 + threadIdx.x;
  if (i < n) p[i] = 0u;
}

// ---- Phase 1: dual GEMM via V_WMMA_F32_16X16X4_F32 -------------------------
// Block: 256 threads = 8 waves; each wave computes a 16x64 tile of x_l and x_r.
// W_l / W_r staged row-major in dynamic LDS (TDM path: offsets 0 and 16384).
__global__ __launch_bounds__(256) void k_gemm_xlr(
    const float* __restrict__ x, const float* __restrict__ Wl,
    const float* __restrict__ Wr, float* __restrict__ xl,
    float* __restrict__ xr, int N) {
  extern __shared__ float lds[];  // [0,4096): W_l  [4096,8192): W_r
  float* WA = lds;
  float* WB = lds + 4096;

#if USE_TDM
  if (threadIdx.x < 32) {               // wave 0 issues both DMAs
    tdm_load_w(Wl, 0u);
    tdm_load_w(Wr, 16384u);
    __builtin_amdgcn_s_wait_tensorcnt(0);
  }
  __syncthreads();
#else
  for (int i = threadIdx.x; i < 4096; i += 256) {
    WA[i] = Wl[i];
    WB[i] = Wr[i];
  }
  __syncthreads();
#endif

  const int wave = threadIdx.x >> 5;
  const int lane = threadIdx.x & 31;
  const int half = lane >> 4;  // 0 or 1 (K-pair select for A/B fragments)
  const int l16  = lane & 15;  // A: row   B/D: column
  const int row0 = (blockIdx.x * 8 + wave) * 16;

  int arow = row0 + l16;
  if (arow >= N) arow = N - 1;  // clamp loads; stores are guarded
  const float* ap = x + (size_t)arow * 64 + half * 2;

  v8f accl[4] = {};
  v8f accr[4] = {};
#pragma unroll
  for (int kt = 0; kt < 16; ++kt) {
    const int kb = kt * 4 + half * 2;   // this lane's K pair
    v2f a = *(const v2f*)(ap + kt * 4);
#pragma unroll
    for (int c = 0; c < 4; ++c) {
      const int col = c * 16 + l16;
      v2f bl = {WA[(kb + 0) * 64 + col], WA[(kb + 1) * 64 + col]};
      v2f br = {WB[(kb + 0) * 64 + col], WB[(kb + 1) * 64 + col]};
      accl[c] = __builtin_amdgcn_wmma_f32_16x16x4_f32(
          false, a, false, bl, (short)0, accl[c], false, false);
      accr[c] = __builtin_amdgcn_wmma_f32_16x16x4_f32(
          false, a, false, br, (short)0, accr[c], false, false);
    }
  }

  // D layout: VGPR r -> M = 8*half + r, N = l16 (16x16 f32 C/D layout)
#pragma unroll
  for (int r = 0; r < 8; ++r) {
    const int row = row0 + half * 8 + r;
    if (row < N) {
      float* pl = xl + (size_t)row * 64 + l16;
      float* pr = xr + (size_t)row * 64 + l16;
#pragma unroll
      for (int c = 0; c < 4; ++c) {
        pl[c * 16] = accl[c][r];
        pr[c * 16] = accr[c][r];
      }
    }
  }
}

// ---- Phase 2: per-edge logits + segment max --------------------------------
__global__ __launch_bounds__(256) void k_edge_logits(
    const int* __restrict__ ei, const float* __restrict__ xl,
    const float* __restrict__ xr, const float* __restrict__ att,
    float* __restrict__ e_out, unsigned* __restrict__ emax, int E, int M) {
  int m = blockIdx.x * 256 + threadIdx.x;
  if (m >= M) return;
  int src, dst;
  if (m < E) { src = ei[m]; dst = ei[E + m]; }
  else       { src = m - E; dst = m - E; }  // self-loop

  const float4* pl = (const float4*)(xl + (size_t)src * 64);
  const float4* pr = (const float4*)(xr + (size_t)dst * 64);
  const float4* pa = (const float4*)att;
  float e = 0.f;
#pragma unroll
  for (int q = 0; q < 16; ++q) {
    float4 a = pl[q], b = pr[q], w = pa[q];
    float s;
    s = a.x + b.x; e += w.x * (s > 0.f ? s : NEG_SLOPE * s);
    s = a.y + b.y; e += w.y * (s > 0.f ? s : NEG_SLOPE * s);
    s = a.z + b.z; e += w.z * (s > 0.f ? s : NEG_SLOPE * s);
    s = a.w + b.w; e += w.w * (s > 0.f ? s : NEG_SLOPE * s);
  }
  e_out[m] = e;
  atomicMax(emax + dst, f2ord(e));
}

// ---- Phase 3: exp-weight + unnormalized scatter-add (wave per edge) --------
__global__ __launch_bounds__(256) void k_edge_accum(
    const int* __restrict__ ei, const float* __restrict__ xl,
    const float* __restrict__ e_arr, const unsigned* __restrict__ emax,
    float* __restrict__ denom, float* __restrict__ out, int E, int M) {
  int m = (blockIdx.x * 256 + threadIdx.x) >> 5;  // edge id (wave per edge)
  int lane = threadIdx.x & 31;
  if (m >= M) return;
  int src, dst;
  if (m < E) { src = ei[m]; dst = ei[E + m]; }
  else       { src = m - E; dst = m - E; }

  float w = __expf(e_arr[m] - ord2f(emax[dst]));
  if (lane == 0) atomicAdd(denom + dst, w);

  const float2 v = *(const float2*)(xl + (size_t)src * 64 + lane * 2);
  float* po = out + (size_t)dst * 64 + lane * 2;
  atomicAdd(po + 0, w * v.x);
  atomicAdd(po + 1, w * v.y);
}

// ---- Phase 4: normalize + bias ---------------------------------------------
__global__ __launch_bounds__(256) void k_finalize(
    float* __restrict__ out, const float* __restrict__ denom,
    const float* __restrict__ bias, int total) {
  int i = blockIdx.x * 256 + threadIdx.x;
  if (i >= total) return;
  out[i] = out[i] / denom[i >> 6] + bias[i & 63];
}

extern "C" void kernel_launch(void* const* d_in, const int* in_sizes, int n_in,
                              void* d_out, int out_size, void* d_ws,
                              size_t ws_size, hipStream_t stream) {
  const float* x    = (const float*)d_in[0];
  const int*   ei   = (const int*)d_in[1];   // [2,E] int32 (JAX x64 disabled)
  const float* Wl   = (const float*)d_in[2];
  const float* Wr   = (const float*)d_in[3];
  const float* att  = (const float*)d_in[4];
  const float* bias = (const float*)d_in[5];

  const int N = in_sizes[0] / 64;
  const int E = in_sizes[1] / 2;
  const int M = E + N;

  // workspace carve (all 4B-typed, rows 256B so float4 access stays aligned)
  float*    xl    = (float*)d_ws;
  float*    xr    = xl + (size_t)N * 64;
  float*    e_arr = xr + (size_t)N * 64;
  unsigned* emax  = (unsigned*)(e_arr + M);
  float*    denom = (float*)(emax + N);

  float* out = (float*)d_out;
  const int nz = N * 64;

  k_zero_u32<<<(nz + 255) / 256, 256, 0, stream>>>((unsigned*)out, nz);
  k_zero_u32<<<(N + 255) / 256, 256, 0, stream>>>(emax, N);
  k_zero_u32<<<(N + 255) / 256, 256, 0, stream>>>((unsigned*)denom, N);

  k_gemm_xlr<<<(N + 127) / 128, 256, 32768, stream>>>(x, Wl, Wr, xl, xr, N);
  k_edge_logits<<<(M + 255) / 256, 256, 0, stream>>>(ei, xl, xr, att, e_arr,
                                                     emax, E, M);
  k_edge_accum<<<(M + 7) / 8, 256, 0, stream>>>(ei, xl, e_arr, emax, denom,
                                                out, E, M);
  k_finalize<<<(nz + 255) / 256, 256, 0, stream>>>(out, denom, bias, nz);
}